// SenseEmbedding_12421045420636
// MI455X (gfx1250) — compile-verified
//
#include <hip/hip_runtime.h>
#include <math.h>

typedef __attribute__((ext_vector_type(2))) float v2f;
typedef __attribute__((ext_vector_type(8))) float v8f;

#define VOCAB 50000
#define D     300
#define NS    5
#define CTX   10
#define NB    16384

#define TILE      16   // batch rows per wave (WMMA M/N)
#define WAVES     2    // waves per block
#define CT_STRIDE 302  // LDS row stride in floats: even (8B-aligned pairs), 302%64=46 -> conflict-light

// ---------------------------------------------------------------------------
// Kernel 0: zero the 5 partial sums living in scratch
// ---------------------------------------------------------------------------
__global__ void sense_zero(float* __restrict__ partial) {
    if (threadIdx.x < NS) partial[threadIdx.x] = 0.0f;
}

// ---------------------------------------------------------------------------
// Kernel 1: per-wave tile of 16 batch rows.
//  Phase A: gather-sum 10 context rows of W_g into an LDS context tile.
//  Phase B: 75 K-chunks x 5 senses of v_wmma_f32_16x16x4_f32; the diagonal of
//           each 16x16 accumulator is score[s, b_m]; reduce & atomicAdd.
// ---------------------------------------------------------------------------
__global__ __launch_bounds__(WAVES * 32) void sense_main(
    const int*   __restrict__ x,    // (NB, 2+CTX) int32
    const float* __restrict__ Wg,   // (VOCAB, D)
    const float* __restrict__ Ws,   // (VOCAB, NS, D)
    float*       __restrict__ partial)  // (NS,) partial logits
{
    __shared__ float Ct[WAVES][TILE * CT_STRIDE];

    const int lane = threadIdx.x & 31;
    const int wid  = threadIdx.x >> 5;
    const int tile = blockIdx.x * WAVES + wid;
    const int b0   = tile * TILE;
    float* ct = Ct[wid];

    // ---------------- Phase A: context sums into LDS ----------------
    for (int bi = 0; bi < TILE; ++bi) {
        const int* xr = x + (long)(b0 + bi) * (2 + CTX);
        int idx[CTX];
#pragma unroll
        for (int j = 0; j < CTX; ++j) idx[j] = xr[2 + j];

#pragma unroll
        for (int db = 0; db < (D + 31) / 32; ++db) {
            int d = db * 32 + lane;
            if (d < D) {
                float acc = 0.0f;
#pragma unroll
                for (int j = 0; j < CTX; ++j)
                    acc += Wg[(long)idx[j] * D + d];   // coalesced 32-lane row read
                ct[bi * CT_STRIDE + d] = acc;
            }
        }
    }
    __syncthreads();

    // ---------------- Phase B: WMMA scores ----------------
    // A (16x4 f32): lane<16 -> (M=lane, K={0,1}); lane>=16 -> (M=lane-16, K={2,3})
    // B (4x16 f32): lane<16 -> (N=lane, K={0,1}); lane>=16 -> (N=lane-16, K={2,3})
    const int half = lane >> 4;       // 0 or 1 -> K pair {0,1} vs {2,3}
    const int mn   = lane & 15;       // M (A row) and N (B col) = batch row in tile

    const long srow = (long)x[(long)(b0 + mn) * (2 + CTX)];   // x[b,0]
    const float* wrow = Ws + srow * (NS * D);                  // this lane's sense block
    const float* arow = ct + mn * CT_STRIDE + 2 * half;

    v8f acc[NS];
#pragma unroll
    for (int s = 0; s < NS; ++s) acc[s] = (v8f){};

    for (int kc = 0; kc < D / 4; ++kc) {           // 75 K-chunks of 4
        const int kb = kc * 4;
        v2f a = *(const v2f*)(arow + kb);          // ds_load_b64 from context tile
#pragma unroll
        for (int s = 0; s < NS; ++s) {
            v2f bv = *(const v2f*)(wrow + s * D + kb + 2 * half);  // global b64 gather
            acc[s] = __builtin_amdgcn_wmma_f32_16x16x4_f32(
                false, a, false, bv, (short)0, acc[s], false, false);
        }
    }

    // Diagonal extraction per the 16x16 f32 C/D layout:
    //   lanes 0-15:  (M=r,   N=lane)     -> diag at r == lane       (lanes 0-7)
    //   lanes 16-31: (M=r+8, N=lane-16)  -> diag at r == lane - 24  (lanes 24-31)
#pragma unroll
    for (int s = 0; s < NS; ++s) {
        float dsum = 0.0f;
#pragma unroll
        for (int r = 0; r < 8; ++r) {
            float el = acc[s][r];
            dsum += ((lane == r) || (lane == 24 + r)) ? el : 0.0f;
        }
#pragma unroll
        for (int off = 16; off > 0; off >>= 1)
            dsum += __shfl_xor(dsum, off, 32);
        if (lane == 0) atomicAdd(&partial[s], dsum);   // global_atomic_add_f32
    }
}

// ---------------------------------------------------------------------------
// Kernel 2: sigmoid of the 5 accumulated logits
// ---------------------------------------------------------------------------
__global__ void sense_sigmoid(const float* __restrict__ partial,
                              float* __restrict__ out) {
    if (threadIdx.x < NS) {
        float v = partial[threadIdx.x];
        out[threadIdx.x] = 1.0f / (1.0f + expf(-v));
    }
}

// ---------------------------------------------------------------------------
extern "C" void kernel_launch(void* const* d_in, const int* in_sizes, int n_in,
                              void* d_out, int out_size, void* d_ws, size_t ws_size,
                              hipStream_t stream) {
    const int*   x  = (const int*)  d_in[0];
    const float* Wg = (const float*)d_in[1];
    const float* Ws = (const float*)d_in[2];
    float* out      = (float*)d_out;
    float* partial  = (float*)d_ws;   // 5 floats of scratch

    sense_zero<<<1, 32, 0, stream>>>(partial);

    const int blocks = NB / (TILE * WAVES);   // 16384 / 32 = 512
    sense_main<<<blocks, WAVES * 32, 0, stream>>>(x, Wg, Ws, partial);

    sense_sigmoid<<<1, 32, 0, stream>>>(partial, out);
}